// ProtoSamNet_3324304687659
// MI455X (gfx1250) — compile-verified
//
#include <hip/hip_runtime.h>
#include <math.h>

#define NPIX   262144
#define CDIM   434
#define CPAD   448
#define KCLS   14
#define MPROT  10
#define KM     140          // K*M
#define KMPAD  144
#define GAMMA_EMA 0.999f
#define INV_EPS_SINK 20.0f  // 1/0.05
#define LN_EPS 1e-5f

typedef float v2f __attribute__((ext_vector_type(2)));
typedef float v8f __attribute__((ext_vector_type(8)));

// ---------- reductions (wave32) ----------
__device__ __forceinline__ float waveReduceSum(float v) {
#pragma unroll
  for (int o = 16; o > 0; o >>= 1) v += __shfl_xor(v, o, 32);
  return v;
}

__device__ float blockReduceSum(float v) {
  __shared__ float ssum[8];
  __shared__ float bres;
  v = waveReduceSum(v);
  int lane = threadIdx.x & 31, w = threadIdx.x >> 5;
  if (lane == 0) ssum[w] = v;
  __syncthreads();
  if (w == 0) {
    float t = (threadIdx.x < (blockDim.x >> 5)) ? ssum[lane] : 0.f;
    t = waveReduceSum(t);
    if (lane == 0) bres = t;
  }
  __syncthreads();
  float r = bres;
  __syncthreads();
  return r;
}

// ---------- zero small accumulator tables ----------
__global__ void zero_tbl_kernel(float* tbl) {
  for (int i = threadIdx.x; i < 1024; i += blockDim.x) tbl[i] = 0.f;
}

// ---------- l2-normalize prototypes into padded [KMPAD, CPAD] ----------
__global__ void protos_norm_kernel(const float* __restrict__ protos,
                                   float* __restrict__ pbuf) {
  int row = blockIdx.x;
  float* out = pbuf + (size_t)row * CPAD;
  if (row >= KM) {
    for (int c = threadIdx.x; c < CPAD; c += blockDim.x) out[c] = 0.f;
    return;
  }
  const float* in = protos + (size_t)row * CDIM;
  float ss = 0.f;
  for (int c = threadIdx.x; c < CDIM; c += blockDim.x) { float v = in[c]; ss += v * v; }
  ss = blockReduceSum(ss);
  float inv = 1.f / fmaxf(sqrtf(ss), 1e-12f);
  for (int c = threadIdx.x; c < CPAD; c += blockDim.x)
    out[c] = (c < CDIM) ? in[c] * inv : 0.f;
}

// ---------- fused LayerNorm + l2norm of feat -> cbuf [NPIX, CPAD] ----------
__global__ void feat_norm_kernel(const float* __restrict__ feat,
                                 const float* __restrict__ g,
                                 const float* __restrict__ b,
                                 float* __restrict__ cbuf) {
  int wave = threadIdx.x >> 5, lane = threadIdx.x & 31;
  int row = blockIdx.x * (blockDim.x >> 5) + wave;
  const float* in = feat + (size_t)row * CDIM;
  float x[14];
  float s = 0.f, s2 = 0.f;
#pragma unroll
  for (int j = 0; j < 14; ++j) {
    int i = lane + 32 * j;
    float v = (i < CDIM) ? in[i] : 0.f;
    x[j] = v; s += v; s2 += v * v;
  }
  s = waveReduceSum(s);
  s2 = waveReduceSum(s2);
  float mu = s * (1.f / CDIM);
  float var = s2 * (1.f / CDIM) - mu * mu;
  float rs = rsqrtf(var + LN_EPS);
  float y[14];
  float n2 = 0.f;
#pragma unroll
  for (int j = 0; j < 14; ++j) {
    int i = lane + 32 * j;
    float yy = 0.f;
    if (i < CDIM) yy = (x[j] - mu) * rs * g[i] + b[i];
    y[j] = yy; n2 += yy * yy;
  }
  n2 = waveReduceSum(n2);
  float inv = 1.f / fmaxf(sqrtf(n2), 1e-12f);
  float* out = cbuf + (size_t)row * CPAD;
#pragma unroll
  for (int j = 0; j < 14; ++j) {
    int i = lane + 32 * j;
    out[i] = y[j] * inv;   // pads (i>=434) write 0
  }
}

// ---------- WMMA fp32 GEMM: logits[N, KM] = cbuf @ pbuf^T ----------
// One wave computes a 16x16 tile using V_WMMA_F32_16X16X4_F32.
// A (16x4 f32): lanes 0-15 M=0..15 (K=0,1), lanes 16-31 same M (K=2,3).
// B (4x16 f32): lanes 0-15 N=0..15 (K=0,1), lanes 16-31 same N (K=2,3).
// C/D (16x16): VGPR r: lanes 0-15 -> M=r, lanes 16-31 -> M=r+8; N=lane&15.
__global__ void gemm_wmma_kernel(const float* __restrict__ cbuf,
                                 const float* __restrict__ pbuf,
                                 float* __restrict__ logits) {
  int wave = threadIdx.x >> 5;
  int lane = threadIdx.x & 31;
  int tile = blockIdx.x * 4 + wave;       // 147456 tiles total
  int rowTile = tile / 9;                 // 9 column tiles (144 padded cols)
  int colTile = tile - rowTile * 9;
  int n0 = rowTile << 4;
  int j0 = colTile << 4;
  int half = lane >> 4;                   // 0 or 1
  int khalf = half << 1;                  // 0 or 2
  int mrow = lane & 15;
  const float* aptr = cbuf + (size_t)(n0 + mrow) * CPAD + khalf;
  const float* bptr = pbuf + (size_t)(j0 + mrow) * CPAD + khalf;
  v8f acc = {};
#pragma unroll 4
  for (int kk = 0; kk < 436; kk += 4) {   // pads beyond 434 are zero
    v2f a = *(const v2f*)(aptr + kk);
    v2f bv = *(const v2f*)(bptr + kk);
    acc = __builtin_amdgcn_wmma_f32_16x16x4_f32(
        false, a, false, bv, (short)0, acc, false, false);
  }
  int col = j0 + mrow;
  if (col < KM) {
    int rbase = n0 + (half << 3);
#pragma unroll
    for (int r = 0; r < 8; ++r)
      logits[(size_t)(rbase + r) * KM + col] = acc[r];
  }
}

// ---------- out_seg = LN_K(max_M sim), pred, correct ----------
__global__ void outseg_kernel(const float* __restrict__ logits,
                              const float* __restrict__ mg,
                              const float* __restrict__ mb,
                              const int* __restrict__ gt,
                              float* __restrict__ out_seg,
                              float* __restrict__ correct) {
  int n = blockIdx.x * blockDim.x + threadIdx.x;
  const float* lr = logits + (size_t)n * KM;
  float s[KCLS];
  float sum = 0.f;
#pragma unroll
  for (int k = 0; k < KCLS; ++k) {
    float mx = -3.402823466e38f;
#pragma unroll
    for (int m = 0; m < MPROT; ++m) mx = fmaxf(mx, lr[k * MPROT + m]);
    s[k] = mx; sum += mx;
  }
  float mu = sum * (1.f / KCLS);
  float v2 = 0.f;
#pragma unroll
  for (int k = 0; k < KCLS; ++k) { float d = s[k] - mu; v2 += d * d; }
  float rs = rsqrtf(v2 * (1.f / KCLS) + LN_EPS);
  int pred = 0;
  float best = -3.402823466e38f;
#pragma unroll
  for (int k = 0; k < KCLS; ++k) {
    float y = (s[k] - mu) * rs * mg[k] + mb[k];
    out_seg[(size_t)n * KCLS + k] = y;
    if (y > best) { best = y; pred = k; }
  }
  correct[n] = (gt[n] == pred) ? 1.f : 0.f;
}

// ---------- sinkhorn pass 1: A1[k,m] = sum_n exp(s/eps), B[k] = count ----------
__global__ void sink_reduce1(const float* __restrict__ logits,
                             const int* __restrict__ gt,
                             float* __restrict__ A1, float* __restrict__ Barr) {
  __shared__ float sA[KM];
  __shared__ float sB[KCLS];
  int t = threadIdx.x;
  if (t < KM) sA[t] = 0.f;
  if (t < KCLS) sB[t] = 0.f;
  __syncthreads();
  int n = blockIdx.x * blockDim.x + t;
  int k = gt[n];
  const float* lr = logits + (size_t)n * KM + k * MPROT;
#pragma unroll
  for (int m = 0; m < MPROT; ++m)
    atomicAdd(&sA[k * MPROT + m], __expf(lr[m] * INV_EPS_SINK));
  atomicAdd(&sB[k], 1.f);
  __syncthreads();
  if (t < KM) atomicAdd(&A1[t], sA[t]);
  if (t < KCLS) atomicAdd(&Barr[t], sB[t]);
}

// tbl layout: A1@0 A2@140 A3@280 c1@420 c2@560 c3@700 S@840 B@854 ncnt@868
__global__ void sink_scalar1(float* tbl) {
  float* A1 = tbl; float* c1 = tbl + 420; float* S = tbl + 840;
  int t = threadIdx.x;
  if (t < KCLS) {
    float s = 0.f;
    for (int m = 0; m < MPROT; ++m) s += A1[t * MPROT + m];
    S[t] = s;
  }
  __syncthreads();
  if (t < KM) {
    int k = t / MPROT;
    float colsum = A1[t] / fmaxf(S[k], 1e-12f);
    c1[t] = 1.f / (fmaxf(colsum, 1e-12f) * (float)MPROT);
  }
}

__global__ void sink_reduce2(const float* __restrict__ logits,
                             const int* __restrict__ gt,
                             const float* __restrict__ tbl,
                             float* __restrict__ A2) {
  const float* c1 = tbl + 420; const float* S = tbl + 840; const float* Barr = tbl + 854;
  __shared__ float sA[KM];
  int t = threadIdx.x;
  if (t < KM) sA[t] = 0.f;
  __syncthreads();
  int n = blockIdx.x * blockDim.x + t;
  int k = gt[n];
  const float* lr = logits + (size_t)n * KM + k * MPROT;
  float e[MPROT];
  float R1 = 0.f;
#pragma unroll
  for (int m = 0; m < MPROT; ++m) {
    e[m] = __expf(lr[m] * INV_EPS_SINK);
    R1 += e[m] * c1[k * MPROT + m];
  }
  float r0 = 1.f / fmaxf(S[k], 1e-12f);
  float Bm = fmaxf(Barr[k], 1.f);
  float r1 = r0 / (fmaxf(r0 * R1, 1e-12f) * Bm);
#pragma unroll
  for (int m = 0; m < MPROT; ++m) atomicAdd(&sA[k * MPROT + m], e[m] * r1);
  __syncthreads();
  if (t < KM) atomicAdd(&A2[t], sA[t]);
}

__global__ void sink_scalar2(float* tbl) {
  float* A2 = tbl + 140; float* c1 = tbl + 420; float* c2 = tbl + 560;
  int t = threadIdx.x;
  if (t < KM) c2[t] = c1[t] / (fmaxf(c1[t] * A2[t], 1e-12f) * (float)MPROT);
}

__global__ void sink_reduce3(const float* __restrict__ logits,
                             const int* __restrict__ gt,
                             const float* __restrict__ tbl,
                             float* __restrict__ A3) {
  const float* c1 = tbl + 420; const float* c2 = tbl + 560;
  const float* S = tbl + 840; const float* Barr = tbl + 854;
  __shared__ float sA[KM];
  int t = threadIdx.x;
  if (t < KM) sA[t] = 0.f;
  __syncthreads();
  int n = blockIdx.x * blockDim.x + t;
  int k = gt[n];
  const float* lr = logits + (size_t)n * KM + k * MPROT;
  float e[MPROT];
  float R1 = 0.f, R2 = 0.f;
#pragma unroll
  for (int m = 0; m < MPROT; ++m) {
    e[m] = __expf(lr[m] * INV_EPS_SINK);
    R1 += e[m] * c1[k * MPROT + m];
    R2 += e[m] * c2[k * MPROT + m];
  }
  float r0 = 1.f / fmaxf(S[k], 1e-12f);
  float Bm = fmaxf(Barr[k], 1.f);
  float r1 = r0 / (fmaxf(r0 * R1, 1e-12f) * Bm);
  float r2 = r1 / (fmaxf(r1 * R2, 1e-12f) * Bm);
#pragma unroll
  for (int m = 0; m < MPROT; ++m) atomicAdd(&sA[k * MPROT + m], e[m] * r2);
  __syncthreads();
  if (t < KM) atomicAdd(&A3[t], sA[t]);
}

__global__ void sink_scalar3(float* tbl) {
  float* A3 = tbl + 280; float* c2 = tbl + 560; float* c3 = tbl + 700;
  int t = threadIdx.x;
  if (t < KM) c3[t] = c2[t] / (fmaxf(c2[t] * A3[t], 1e-12f) * (float)MPROT);
}

// ---------- hard assignment, proto_target, ncount ----------
__global__ void assign_kernel(const float* __restrict__ logits,
                              const int* __restrict__ gt,
                              const float* __restrict__ tbl,
                              const float* __restrict__ correct,
                              int* __restrict__ target,
                              float* __restrict__ ptarget,
                              float* __restrict__ ncount_dummy /*tbl+868*/) {
  const float* c3 = tbl + 700;
  __shared__ float sN[KM];
  int t = threadIdx.x;
  if (t < KM) sN[t] = 0.f;
  __syncthreads();
  int n = blockIdx.x * blockDim.x + t;
  int k = gt[n];
  const float* lr = logits + (size_t)n * KM + k * MPROT;
  int idx = 0;
  float best = -3.402823466e38f;
#pragma unroll
  for (int m = 0; m < MPROT; ++m) {
    float v = __expf(lr[m] * INV_EPS_SINK) * c3[k * MPROT + m];
    if (v > best) { best = v; idx = m; }
  }
  int km = k * MPROT + idx;
  target[n] = km;
  ptarget[n] = (float)km;
  atomicAdd(&sN[km], correct[n]);
  __syncthreads();
  if (t < KM) atomicAdd(&ncount_dummy[t], sN[t]);
}

// ---------- f[km,:] = sum over pixels assigned to km (correct only) ----------
__global__ void f_scan_kernel(const float* __restrict__ cbuf,
                              const int* __restrict__ target,
                              const float* __restrict__ correct,
                              float* __restrict__ fbuf) {
  __shared__ int stk[256];
  __shared__ float scor[256];
  int b = blockIdx.x;
  int t = threadIdx.x;
  float acc0 = 0.f, acc1 = 0.f;
  for (int base = 0; base < NPIX; base += 256) {
    stk[t] = target[base + t];
    scor[t] = correct[base + t];
    __syncthreads();
    for (int j = 0; j < 256; ++j) {
      if (stk[j] == b && scor[j] > 0.f) {
        const float* row = cbuf + (size_t)(base + j) * CPAD;
        acc0 += row[t];
        if (t + 256 < CDIM) acc1 += row[t + 256];
      }
    }
    __syncthreads();
  }
  fbuf[(size_t)b * CPAD + t] = acc0;
  if (t + 256 < CDIM) fbuf[(size_t)b * CPAD + t + 256] = acc1;
}

// ---------- momentum update + double l2norm -> new_protos ----------
__global__ void finalize_kernel(const float* __restrict__ fbuf,
                                const float* __restrict__ pbuf,
                                const float* __restrict__ tbl,
                                float* __restrict__ newp) {
  int km = blockIdx.x;
  const float* f = fbuf + (size_t)km * CPAD;
  const float* p = pbuf + (size_t)km * CPAD;
  float cnt = tbl[868 + km];
  int t = threadIdx.x;
  float f0 = (t < CDIM) ? f[t] : 0.f;
  float f1 = (t + 256 < CDIM) ? f[t + 256] : 0.f;
  float fss = blockReduceSum(f0 * f0 + f1 * f1);
  float finv = 1.f / fmaxf(sqrtf(fss), 1e-12f);
  float p0 = (t < CDIM) ? p[t] : 0.f;
  float p1 = (t + 256 < CDIM) ? p[t + 256] : 0.f;
  float v0 = (cnt > 0.f) ? (GAMMA_EMA * p0 + (1.f - GAMMA_EMA) * f0 * finv) : p0;
  float v1 = (cnt > 0.f) ? (GAMMA_EMA * p1 + (1.f - GAMMA_EMA) * f1 * finv) : p1;
  float vss = blockReduceSum(v0 * v0 + v1 * v1);
  float vinv = 1.f / fmaxf(sqrtf(vss), 1e-12f);
  float* o = newp + (size_t)km * CDIM;
  if (t < CDIM) o[t] = v0 * vinv;
  if (t + 256 < CDIM) o[t + 256] = v1 * vinv;
}

// ==================== launcher ====================
extern "C" void kernel_launch(void* const* d_in, const int* in_sizes, int n_in,
                              void* d_out, int out_size, void* d_ws, size_t ws_size,
                              hipStream_t stream) {
  const float* feat  = (const float*)d_in[0];
  const float* proto = (const float*)d_in[1];
  const float* fg    = (const float*)d_in[2];
  const float* fb    = (const float*)d_in[3];
  const float* mg    = (const float*)d_in[4];
  const float* mb    = (const float*)d_in[5];
  const int*   gt    = (const int*)d_in[6];

  float* out = (float*)d_out;
  float* out_seg = out;
  float* logits  = out + (size_t)NPIX * KCLS;
  float* ptarget = logits + (size_t)NPIX * KM;
  float* newp    = ptarget + NPIX;

  float* ws   = (float*)d_ws;
  float* cbuf = ws;                                   // [NPIX, CPAD]
  float* pbuf = cbuf + (size_t)NPIX * CPAD;           // [KMPAD, CPAD]
  float* tbl  = pbuf + (size_t)KMPAD * CPAD;          // 1024 scalars
  float* fbuf = tbl + 1024;                           // [KM, CPAD]
  float* corr = fbuf + (size_t)KM * CPAD;             // [NPIX]
  int*   targ = (int*)(corr + NPIX);                  // [NPIX]

  float* A1 = tbl;
  float* A2 = tbl + 140;
  float* A3 = tbl + 280;
  float* Barr = tbl + 854;
  float* ncnt = tbl + 868;

  zero_tbl_kernel<<<1, 256, 0, stream>>>(tbl);
  protos_norm_kernel<<<KMPAD, 256, 0, stream>>>(proto, pbuf);
  feat_norm_kernel<<<NPIX / 8, 256, 0, stream>>>(feat, fg, fb, cbuf);
  gemm_wmma_kernel<<<(NPIX / 16) * 9 / 4, 128, 0, stream>>>(cbuf, pbuf, logits);
  outseg_kernel<<<NPIX / 256, 256, 0, stream>>>(logits, mg, mb, gt, out_seg, corr);
  sink_reduce1<<<NPIX / 256, 256, 0, stream>>>(logits, gt, A1, Barr);
  sink_scalar1<<<1, 256, 0, stream>>>(tbl);
  sink_reduce2<<<NPIX / 256, 256, 0, stream>>>(logits, gt, tbl, A2);
  sink_scalar2<<<1, 256, 0, stream>>>(tbl);
  sink_reduce3<<<NPIX / 256, 256, 0, stream>>>(logits, gt, tbl, A3);
  sink_scalar3<<<1, 256, 0, stream>>>(tbl);
  assign_kernel<<<NPIX / 256, 256, 0, stream>>>(logits, gt, tbl, corr, targ, ptarget, ncnt);
  f_scan_kernel<<<KM, 256, 0, stream>>>(cbuf, targ, corr, fbuf);
  finalize_kernel<<<KM, 256, 0, stream>>>(fbuf, pbuf, tbl, newp);
}